// OCGatherEnergyCorrFac3_86603720556733
// MI455X (gfx1250) — compile-verified
//
#include <hip/hip_runtime.h>
#include <hip/hip_bf16.h>

// ---------------------------------------------------------------------------
// OCGatherEnergyCorrFac3: segment-sum of corrected hit energies + gather-back.
//   acc[sid+1] += (is_track[idx]==0 ? energy[idx] : 0) * (sid==-1 ? 0 : corr)
//   out[i]      = acc[sid[i]+1]
// Memory-bound: ~160 MB working set, fully L2-resident (192 MB L2 on MI455X);
// HBM floor ~7 us @ 23.3 TB/s. No matmul structure -> WMMA not applicable.
// CDNA5-specific paths used instead:
//   * global_load_async_to_lds_b128 + s_wait_asynccnt (ASYNCcnt engine) to
//     stage the sid stream through LDS, overlapped with the random gathers.
//   * global_prefetch_b8 (__builtin_prefetch) to warm the 80 KB accumulator
//     table into WGP-local cache before the 6M random gathers in pass 2.
// ---------------------------------------------------------------------------

#define NSEG 20001  // N_SHOWERS + 1 (fixed by the reference)
#define TB   256

__global__ void ocg_zero_kernel(float* __restrict__ acc, int n) {
    int i = blockIdx.x * blockDim.x + threadIdx.x;
    if (i < n) acc[i] = 0.0f;
}

// One thread handles 4 consecutive hits. sid quad is fetched via the CDNA5
// async-to-LDS engine (per-lane global + LDS addresses); the copy is in
// flight while the corr/idx b128 loads and all 8 random gathers issue.
__global__ void ocg_accum_kernel(const int*   __restrict__ sid,
                                 const float* __restrict__ corr,
                                 const int*   __restrict__ idx,
                                 const float* __restrict__ energy,
                                 const int*   __restrict__ is_track,
                                 float*       __restrict__ acc,
                                 int n) {
    __shared__ int4 s_sid[TB];
    const int tid  = threadIdx.x;
    const int base = (blockIdx.x * blockDim.x + tid) * 4;

    if (base + 3 < n) {
        // Kick off async global->LDS copy of this lane's sid quad (16B).
        const int4* gsid = reinterpret_cast<const int4*>(sid + base);
        unsigned lds_addr = (unsigned)(unsigned long long)&s_sid[tid];
        asm volatile("global_load_async_to_lds_b128 %0, %1, off"
                     :: "v"(lds_addr), "v"(gsid)
                     : "memory");

        // Overlap: streaming b128 loads + 8 random gathers (all in flight).
        float4 c = *reinterpret_cast<const float4*>(corr + base);
        int4   x = *reinterpret_cast<const int4*>(idx + base);
        int   t0 = is_track[x.x], t1 = is_track[x.y],
              t2 = is_track[x.z], t3 = is_track[x.w];
        float e0 = energy[x.x], e1 = energy[x.y],
              e2 = energy[x.z], e3 = energy[x.w];

        // Drain the async engine, then read sid back from LDS.
        asm volatile("s_wait_asynccnt 0x0" ::: "memory");
        int4 s = s_sid[tid];

        if (s.x != -1 && t0 == 0) atomicAdd(&acc[s.x + 1], e0 * c.x);
        if (s.y != -1 && t1 == 0) atomicAdd(&acc[s.y + 1], e1 * c.y);
        if (s.z != -1 && t2 == 0) atomicAdd(&acc[s.z + 1], e2 * c.z);
        if (s.w != -1 && t3 == 0) atomicAdd(&acc[s.w + 1], e3 * c.w);
    } else {
        for (int j = base; j < n; ++j) {
            int sj = sid[j];
            int xj = idx[j];
            if (sj != -1 && is_track[xj] == 0)
                atomicAdd(&acc[sj + 1], energy[xj] * corr[j]);
        }
    }
}

// out[i] = acc[sid[i]+1]. Warm the 80 KB acc table into WGP-local cache
// first (global_prefetch_b8, ~3 cache lines per thread), then b128 in/out.
__global__ void ocg_gather_kernel(const int*   __restrict__ sid,
                                  const float* __restrict__ acc,
                                  float*       __restrict__ out,
                                  int n) {
    const char* accb = (const char*)acc;
    for (int off = threadIdx.x * 128; off < NSEG * 4; off += TB * 128)
        __builtin_prefetch(accb + off, 0, 3);

    int base = (blockIdx.x * blockDim.x + threadIdx.x) * 4;
    if (base + 3 < n) {
        int4 s = *reinterpret_cast<const int4*>(sid + base);
        float4 o;
        o.x = acc[s.x + 1];
        o.y = acc[s.y + 1];
        o.z = acc[s.z + 1];
        o.w = acc[s.w + 1];
        *reinterpret_cast<float4*>(out + base) = o;
    } else {
        for (int j = base; j < n; ++j) out[j] = acc[sid[j] + 1];
    }
}

extern "C" void kernel_launch(void* const* d_in, const int* in_sizes, int n_in,
                              void* d_out, int out_size, void* d_ws, size_t ws_size,
                              hipStream_t stream) {
    // setup_inputs() order:
    //   0: pred_sid (i32), 1: pred_corr_factor (f32), 2: rechit_energy (f32),
    //   3: no_noise_idx (i32), 4: pred_beta (f32, unused), 5: is_track (i32)
    const int*   sid      = (const int*)  d_in[0];
    const float* corr     = (const float*)d_in[1];
    const float* energy   = (const float*)d_in[2];
    const int*   idx      = (const int*)  d_in[3];
    const int*   is_track = (const int*)  d_in[5];
    float*       out      = (float*)d_out;
    float*       acc      = (float*)d_ws;   // NSEG floats = 80,004 bytes
    (void)n_in; (void)ws_size; (void)out_size;

    const int n = in_sizes[0];              // N_FILT

    ocg_zero_kernel<<<(NSEG + TB - 1) / TB, TB, 0, stream>>>(acc, NSEG);

    const int quads_per_block = TB * 4;
    const int blocks = (n + quads_per_block - 1) / quads_per_block;
    ocg_accum_kernel<<<blocks, TB, 0, stream>>>(sid, corr, idx, energy,
                                                is_track, acc, n);
    ocg_gather_kernel<<<blocks, TB, 0, stream>>>(sid, acc, out, n);
}